// DotPredictor_26319559590591
// MI455X (gfx1250) — compile-verified
//
#include <hip/hip_runtime.h>
#include <hip/hip_bf16.h>

typedef float v2f __attribute__((ext_vector_type(2)));
typedef float v8f __attribute__((ext_vector_type(8)));

#define D_FEAT 128
#define EDGES_PER_WAVE 16
#define WAVES_PER_BLOCK 8

// One wave32 handles 16 edges. Edge r's src row -> A-matrix row r, its dst row
// -> B-matrix column r; 32 chained V_WMMA_F32_16X16X4_F32 ops accumulate over
// the 128 features, and D[r][r] is the per-edge dot product.
//
// f32 WMMA fragment layouts (CDNA5 ISA 7.12.2):
//   A[m][k]: lane = m + 16*(k>>1), vgpr = k&1   (16x4)
//   B[k][n]: lane = n + 16*(k>>1), vgpr = k&1   (4x16, mirrored)
// => for step i, lane L loads float2 at row[e_{L&15}] + 4*i + 2*(L>>4)
//    identically for A (src row) and B (dst row). No shuffles, no LDS.
__global__ __launch_bounds__(WAVES_PER_BLOCK * 32)
void edge_dot_wmma(const float* __restrict__ h,
                   const int*  __restrict__ src,
                   const int*  __restrict__ dst,
                   float* __restrict__ out,
                   int nEdges)
{
    const int  lane = threadIdx.x & 31;
    const long tile = (long)blockIdx.x * WAVES_PER_BLOCK + (threadIdx.x >> 5);
    const long base = tile * EDGES_PER_WAVE;
    if (base >= nEdges) return;              // wave-uniform: EXEC stays all-1s

    const int r    = lane & 15;              // edge slot within the 16-edge tile
    const int half = lane >> 4;              // selects K-half {0,1} -> k = 2*half + j

    long e = base + r;
    e = (e < (long)nEdges) ? e : base;       // clamp tail (cndmask, not branch)

    const long sRow = (long)src[e];
    const long dRow = (long)dst[e];
    const float* pa = h + sRow * D_FEAT + half * 2;   // A fragment source (src row)
    const float* pb = h + dRow * D_FEAT + half * 2;   // B fragment source (dst row)

    v8f acc = {0.f, 0.f, 0.f, 0.f, 0.f, 0.f, 0.f, 0.f};
#pragma unroll 8
    for (int i = 0; i < 32; ++i) {           // 32 * K=4 = 128 features
        v2f a = *(const v2f*)(pa + 4 * i);   // global_load_b64
        v2f b = *(const v2f*)(pb + 4 * i);   // global_load_b64
        acc = __builtin_amdgcn_wmma_f32_16x16x4_f32(
            /*neg_a=*/false, a, /*neg_b=*/false, b,
            /*c_mod=*/(short)0, acc, /*reuse_a=*/false, /*reuse_b=*/false);
    }

    // Diagonal D[q][q]: q<8 -> lane q, acc[q]; q>=8 -> lane q+16, acc[q-8].
    const bool writer = (lane < 8) || (lane >= 24);
    const int  q = (lane < 8) ? lane : (lane - 16);   // edge slot owned by lane
    const int  v = q & 7;                             // acc element index
    float val = acc[0];
    val = (v == 1) ? acc[1] : val;
    val = (v == 2) ? acc[2] : val;
    val = (v == 3) ? acc[3] : val;
    val = (v == 4) ? acc[4] : val;
    val = (v == 5) ? acc[5] : val;
    val = (v == 6) ? acc[6] : val;
    val = (v == 7) ? acc[7] : val;

    const long oe = base + q;
    if (writer && oe < (long)nEdges) out[oe] = val;
}

extern "C" void kernel_launch(void* const* d_in, const int* in_sizes, int n_in,
                              void* d_out, int out_size, void* d_ws, size_t ws_size,
                              hipStream_t stream)
{
    const float* h   = (const float*)d_in[0];
    const int*   src = (const int*)d_in[1];
    const int*   dst = (const int*)d_in[2];
    float*       out = (float*)d_out;
    const int nEdges = in_sizes[1];

    const long tiles  = ((long)nEdges + EDGES_PER_WAVE - 1) / EDGES_PER_WAVE;
    const int  blocks = (int)((tiles + WAVES_PER_BLOCK - 1) / WAVES_PER_BLOCK);
    edge_dot_wmma<<<blocks, WAVES_PER_BLOCK * 32, 0, stream>>>(h, src, dst, out, nEdges);
}